// MatrixMixer_26989574488638
// MI455X (gfx1250) — compile-verified
//
#include <hip/hip_runtime.h>

typedef float v2f __attribute__((ext_vector_type(2)));
typedef float v4f __attribute__((ext_vector_type(4)));
typedef float v8f __attribute__((ext_vector_type(8)));
typedef int   v4i __attribute__((ext_vector_type(4)));

#define B_  4
#define L_  8192
#define D_  1024
#define NH  16
#define QK  64
// W_in rows: [0,1024) = Wv, [1024,2048) = Wq, [2048,3072) = Wk

#if defined(__has_builtin)
#if __has_builtin(__builtin_amdgcn_global_load_async_to_lds_b128) && \
    __has_builtin(__builtin_amdgcn_s_wait_asynccnt)
#define USE_ASYNC_LDS 1
#endif
#endif

__device__ __forceinline__ v8f wmma4(v2f a, v2f b, v8f c) {
  // D = A(16x4 f32) * B(4x16 f32) + C(16x16 f32)
  return __builtin_amdgcn_wmma_f32_16x16x4_f32(false, a, false, b, (short)0, c, false, false);
}
__device__ __forceinline__ v8f vzero8() {
  v8f z = {0.f,0.f,0.f,0.f,0.f,0.f,0.f,0.f};
  return z;
}

#ifdef USE_ASYNC_LDS
typedef __attribute__((address_space(1))) v4i* as1_v4i_p;
typedef __attribute__((address_space(3))) v4i* as3_v4i_p;
// Async copy 16B global -> LDS (per lane). LDS generic pointer's low 32 bits
// are the LDS byte offset.
__device__ __forceinline__ void async_cp16(const void* g, void* l) {
  __builtin_amdgcn_global_load_async_to_lds_b128(
      (as1_v4i_p)(unsigned long long)(g),
      (as3_v4i_p)(unsigned int)(unsigned long long)(l),
      0, 0);
}
#endif

// ---------------------------------------------------------------------------
// Kernel 1: G[b] = U[b]^T @ U[b]   (per-batch 1024x1024, K = 8192)
// grid.x = b*64 + ti*8 + tj ; 128x128 output tile per WG, 8 waves
// ---------------------------------------------------------------------------
__global__ __launch_bounds__(256) void gram_kernel(const float* __restrict__ u,
                                                   float* __restrict__ G) {
  const int id = blockIdx.x;
  const int b  = id >> 6;
  const int ti = (id >> 3) & 7;
  const int tj = id & 7;
  const float* __restrict__ U  = u + (size_t)b * L_ * D_;
  float* __restrict__       Gb = G + (size_t)b * D_ * D_;
#ifdef USE_ASYNC_LDS
  __shared__ float As[2][32][128];
  __shared__ float Bs[2][32][128];
#else
  __shared__ float As[1][32][128];
  __shared__ float Bs[1][32][128];
#endif
  const int tid  = threadIdx.x;
  const int lane = tid & 31;
  const int wid  = tid >> 5;
  const int n16  = lane & 15;
  const int hi   = lane >> 4;
  const int wr = (wid >> 2) * 64;   // wave row base (0 / 64)
  const int wc = (wid & 3) * 32;    // wave col base (0..96)
  const int i0 = ti * 128, j0 = tj * 128;
  v8f acc[4][2];
#pragma unroll
  for (int r = 0; r < 4; ++r)
#pragma unroll
    for (int c = 0; c < 2; ++c) acc[r][c] = vzero8();

#ifdef USE_ASYNC_LDS
  // prologue: async-stage chunk 0
#pragma unroll
  for (int t = 0; t < 4; ++t) {
    int idx = tid + t * 256, k = idx >> 5, c4 = (idx & 31) * 4;
    async_cp16(&U[(size_t)k * D_ + i0 + c4], &As[0][k][c4]);
    async_cp16(&U[(size_t)k * D_ + j0 + c4], &Bs[0][k][c4]);
  }
  int buf = 0;
#endif

  for (int l0 = 0; l0 < L_; l0 += 32) {
#ifdef USE_ASYNC_LDS
    __builtin_amdgcn_s_wait_asynccnt(0);
    __syncthreads();
    if (l0 + 32 < L_) {
#pragma unroll
      for (int t = 0; t < 4; ++t) {
        int idx = tid + t * 256, k = idx >> 5, c4 = (idx & 31) * 4;
        async_cp16(&U[(size_t)(l0 + 32 + k) * D_ + i0 + c4], &As[buf ^ 1][k][c4]);
        async_cp16(&U[(size_t)(l0 + 32 + k) * D_ + j0 + c4], &Bs[buf ^ 1][k][c4]);
      }
    }
#else
    const int buf = 0;
#pragma unroll
    for (int t = 0; t < 4; ++t) {
      int idx = tid + t * 256, k = idx >> 5, c4 = (idx & 31) * 4;
      *(v4f*)&As[0][k][c4] = *(const v4f*)&U[(size_t)(l0 + k) * D_ + i0 + c4];
      *(v4f*)&Bs[0][k][c4] = *(const v4f*)&U[(size_t)(l0 + k) * D_ + j0 + c4];
      if (l0 + 32 < L_) {  // prefetch next chunk (global_prefetch_b8)
        __builtin_prefetch(&U[(size_t)(l0 + 32 + k) * D_ + i0 + c4], 0, 3);
        __builtin_prefetch(&U[(size_t)(l0 + 32 + k) * D_ + j0 + c4], 0, 3);
      }
    }
    __syncthreads();
#endif
#pragma unroll
    for (int k0 = 0; k0 < 32; k0 += 4) {
      v2f af[4], bf[2];
#pragma unroll
      for (int r = 0; r < 4; ++r) {
        int col = wr + r * 16 + n16;
        af[r].x = As[buf][k0 + 2*hi][col];
        af[r].y = As[buf][k0 + 2*hi + 1][col];
      }
#pragma unroll
      for (int c = 0; c < 2; ++c) {
        int col = wc + c * 16 + n16;
        bf[c].x = Bs[buf][k0 + 2*hi][col];
        bf[c].y = Bs[buf][k0 + 2*hi + 1][col];
      }
#pragma unroll
      for (int r = 0; r < 4; ++r)
#pragma unroll
        for (int c = 0; c < 2; ++c)
          acc[r][c] = wmma4(af[r], bf[c], acc[r][c]);
    }
    __syncthreads();
#ifdef USE_ASYNC_LDS
    buf ^= 1;
#endif
  }
#pragma unroll
  for (int r = 0; r < 4; ++r)
#pragma unroll
    for (int c = 0; c < 2; ++c) {
      float* o = &Gb[(size_t)(i0 + wr + r*16 + 8*hi) * D_ + j0 + wc + c*16 + n16];
#pragma unroll
      for (int rr = 0; rr < 8; ++rr) o[(size_t)rr * D_] = acc[r][c][rr];
    }
}

// ---------------------------------------------------------------------------
// Kernel 2: kv[b,h] = Wk_h @ G[b] @ Wv_h^T  (64x64, chained GEMM via LDS)
// grid.x = b*16 + h ; one WG, 8 waves; 4x4 grid of 16x16 tiles, 2 per wave
// ---------------------------------------------------------------------------
__global__ __launch_bounds__(256) void kv_kernel(const float* __restrict__ W,
                                                 const float* __restrict__ G,
                                                 float* __restrict__ kv) {
  const int b = blockIdx.x >> 4;
  const int h = blockIdx.x & 15;
  const float* __restrict__ Gb = G + (size_t)b * D_ * D_;
  const float* __restrict__ Wv = W + (size_t)(h * QK) * D_;
  const float* __restrict__ Wk = W + (size_t)(2048 + h * QK) * D_;
  __shared__ float T[64 * 64];      // T[dc][p]
  const int tid  = threadIdx.x;
  const int lane = tid & 31;
  const int wid  = tid >> 5;
  const int n16  = lane & 15;
  const int hi   = lane >> 4;
  const int trow = wid >> 1;        // 0..3
  const int tc0  = (wid & 1) * 2;   // 0 / 2
  v8f acckv[2];
  acckv[0] = vzero8(); acckv[1] = vzero8();

  for (int d0 = 0; d0 < D_; d0 += 64) {
    // Phase 1: T[dc][p] = sum_e G[d0+dc][e] * Wv[p][e]   (K = 1024)
    v8f aT[2];
    aT[0] = vzero8(); aT[1] = vzero8();
    for (int e0 = 0; e0 < D_; e0 += 4) {
      v2f af  = *(const v2f*)&Gb[(size_t)(d0 + trow*16 + n16) * D_ + e0 + 2*hi];
      v2f bf0 = *(const v2f*)&Wv[(size_t)(tc0*16 + n16) * D_ + e0 + 2*hi];
      v2f bf1 = *(const v2f*)&Wv[(size_t)((tc0+1)*16 + n16) * D_ + e0 + 2*hi];
      aT[0] = wmma4(af, bf0, aT[0]);
      aT[1] = wmma4(af, bf1, aT[1]);
    }
#pragma unroll
    for (int j = 0; j < 2; ++j) {
      float* o = &T[(trow*16 + 8*hi) * 64 + (tc0 + j)*16 + n16];
#pragma unroll
      for (int rr = 0; rr < 8; ++rr) o[rr * 64] = aT[j][rr];
    }
    __syncthreads();
    // Phase 2: kv[n][p] += sum_dc Wk[n][d0+dc] * T[dc][p]   (K = 64)
#pragma unroll
    for (int k0 = 0; k0 < 64; k0 += 4) {
      v2f af2 = *(const v2f*)&Wk[(size_t)(trow*16 + n16) * D_ + d0 + k0 + 2*hi];
      v2f bf[2];
#pragma unroll
      for (int j = 0; j < 2; ++j) {
        bf[j].x = T[(k0 + 2*hi) * 64 + (tc0 + j)*16 + n16];
        bf[j].y = T[(k0 + 2*hi + 1) * 64 + (tc0 + j)*16 + n16];
      }
      acckv[0] = wmma4(af2, bf[0], acckv[0]);
      acckv[1] = wmma4(af2, bf[1], acckv[1]);
    }
    __syncthreads();
  }
  float* kvb = kv + (size_t)(b * 16 + h) * 64 * 64;
#pragma unroll
  for (int j = 0; j < 2; ++j) {
    float* o = &kvb[(trow*16 + 8*hi) * 64 + (tc0 + j)*16 + n16];
#pragma unroll
    for (int rr = 0; rr < 8; ++rr) o[rr * 64] = acckv[j][rr];
  }
}

// ---------------------------------------------------------------------------
// Kernel 3: Mcat[b][e, h*64+p] = sum_n Wq_h[n,e] * kv[b,h][n,p]   (K = 64)
// grid.x = b*128 + h*8 + et ; WG does 128 e-rows x 64 p-cols; wave = 1 row tile
// ---------------------------------------------------------------------------
__global__ __launch_bounds__(256) void mcat_kernel(const float* __restrict__ W,
                                                   const float* __restrict__ kv,
                                                   float* __restrict__ M) {
  const int id = blockIdx.x;
  const int b  = id >> 7;
  const int h  = (id >> 3) & 15;
  const int et = id & 7;
  const int e0 = et * 128;
  const float* __restrict__ Wq  = W  + (size_t)(1024 + h * QK) * D_;
  const float* __restrict__ kvb = kv + (size_t)(b * 16 + h) * 64 * 64;
  float* __restrict__       Mb  = M  + (size_t)b * D_ * D_;
  const int tid  = threadIdx.x;
  const int lane = tid & 31;
  const int wid  = tid >> 5;     // e-row tile 0..7
  const int n16  = lane & 15;
  const int hi   = lane >> 4;
  v8f acc[4];
#pragma unroll
  for (int c = 0; c < 4; ++c) acc[c] = vzero8();
#pragma unroll
  for (int k0 = 0; k0 < 64; k0 += 4) {
    v2f af;
    af.x = Wq[(size_t)(k0 + 2*hi)     * D_ + e0 + wid*16 + n16];
    af.y = Wq[(size_t)(k0 + 2*hi + 1) * D_ + e0 + wid*16 + n16];
#pragma unroll
    for (int c = 0; c < 4; ++c) {
      v2f bf;
      bf.x = kvb[(k0 + 2*hi)     * 64 + c*16 + n16];
      bf.y = kvb[(k0 + 2*hi + 1) * 64 + c*16 + n16];
      acc[c] = wmma4(af, bf, acc[c]);
    }
  }
#pragma unroll
  for (int c = 0; c < 4; ++c) {
    float* o = &Mb[(size_t)(e0 + wid*16 + 8*hi) * D_ + h*QK + c*16 + n16];
#pragma unroll
    for (int rr = 0; rr < 8; ++rr) o[(size_t)rr * D_] = acc[c][rr];
  }
}

// ---------------------------------------------------------------------------
// Kernel 4: Y[b] = U[b] @ Mcat[b]   (8192x1024 x 1024x1024 per batch)
// grid.x = b*512 + lt*8 + ct ; 128x128 tile per WG, K = 1024, LDS staged
// ---------------------------------------------------------------------------
__global__ __launch_bounds__(256) void y_kernel(const float* __restrict__ u,
                                                const float* __restrict__ M,
                                                float* __restrict__ y) {
  const int id = blockIdx.x;
  const int b  = id >> 9;
  const int lt = (id >> 3) & 63;
  const int ct = id & 7;
  const float* __restrict__ U  = u + (size_t)b * L_ * D_;
  const float* __restrict__ Mb = M + (size_t)b * D_ * D_;
  float* __restrict__       Y  = y + (size_t)b * L_ * D_;
#ifdef USE_ASYNC_LDS
  __shared__ float As[2][128][32];  // As[buf][m][k] = U[l0+m][kb+k]
  __shared__ float Bs[2][32][128];  // Bs[buf][k][n] = Mcat[kb+k][c0+n]
#else
  __shared__ float As[1][128][32];
  __shared__ float Bs[1][32][128];
#endif
  const int tid  = threadIdx.x;
  const int lane = tid & 31;
  const int wid  = tid >> 5;
  const int n16  = lane & 15;
  const int hi   = lane >> 4;
  const int wr = (wid >> 2) * 64;
  const int wc = (wid & 3) * 32;
  const int l0 = lt * 128, c0 = ct * 128;
  v8f acc[4][2];
#pragma unroll
  for (int r = 0; r < 4; ++r)
#pragma unroll
    for (int c = 0; c < 2; ++c) acc[r][c] = vzero8();

#ifdef USE_ASYNC_LDS
#pragma unroll
  for (int t = 0; t < 4; ++t) {
    int idx = tid + t * 256;
    int m = idx >> 3,  c4 = (idx & 7) * 4;
    int k = idx >> 5,  b4 = (idx & 31) * 4;
    async_cp16(&U[(size_t)(l0 + m) * D_ + c4], &As[0][m][c4]);
    async_cp16(&Mb[(size_t)k * D_ + c0 + b4], &Bs[0][k][b4]);
  }
  int buf = 0;
#endif

  for (int kb = 0; kb < D_; kb += 32) {
#ifdef USE_ASYNC_LDS
    __builtin_amdgcn_s_wait_asynccnt(0);
    __syncthreads();
    if (kb + 32 < D_) {
#pragma unroll
      for (int t = 0; t < 4; ++t) {
        int idx = tid + t * 256;
        int m = idx >> 3,  c4 = (idx & 7) * 4;
        int k = idx >> 5,  b4 = (idx & 31) * 4;
        async_cp16(&U[(size_t)(l0 + m) * D_ + kb + 32 + c4], &As[buf ^ 1][m][c4]);
        async_cp16(&Mb[(size_t)(kb + 32 + k) * D_ + c0 + b4], &Bs[buf ^ 1][k][b4]);
      }
    }
#else
    const int buf = 0;
#pragma unroll
    for (int t = 0; t < 4; ++t) {
      int idx = tid + t * 256;
      int m = idx >> 3,  c4 = (idx & 7) * 4;
      int k = idx >> 5,  b4 = (idx & 31) * 4;
      *(v4f*)&As[0][m][c4] = *(const v4f*)&U[(size_t)(l0 + m) * D_ + kb + c4];
      *(v4f*)&Bs[0][k][b4] = *(const v4f*)&Mb[(size_t)(kb + k) * D_ + c0 + b4];
      if (kb + 32 < D_) {
        __builtin_prefetch(&U[(size_t)(l0 + m) * D_ + kb + 32 + c4], 0, 3);
        __builtin_prefetch(&Mb[(size_t)(kb + 32 + k) * D_ + c0 + b4], 0, 3);
      }
    }
    __syncthreads();
#endif
#pragma unroll
    for (int k0 = 0; k0 < 32; k0 += 4) {
      v2f af[4], bf[2];
#pragma unroll
      for (int r = 0; r < 4; ++r)
        af[r] = *(const v2f*)&As[buf][wr + r*16 + n16][k0 + 2*hi];
#pragma unroll
      for (int c = 0; c < 2; ++c) {
        bf[c].x = Bs[buf][k0 + 2*hi][wc + c*16 + n16];
        bf[c].y = Bs[buf][k0 + 2*hi + 1][wc + c*16 + n16];
      }
#pragma unroll
      for (int r = 0; r < 4; ++r)
#pragma unroll
        for (int c = 0; c < 2; ++c)
          acc[r][c] = wmma4(af[r], bf[c], acc[r][c]);
    }
    __syncthreads();
#ifdef USE_ASYNC_LDS
    buf ^= 1;
#endif
  }
#pragma unroll
  for (int r = 0; r < 4; ++r)
#pragma unroll
    for (int c = 0; c < 2; ++c) {
      float* o = &Y[(size_t)(l0 + wr + r*16 + 8*hi) * D_ + c0 + wc + c*16 + n16];
#pragma unroll
      for (int rr = 0; rr < 8; ++rr) o[(size_t)rr * D_] = acc[r][c][rr];
    }
}

// ---------------------------------------------------------------------------
extern "C" void kernel_launch(void* const* d_in, const int* in_sizes, int n_in,
                              void* d_out, int out_size, void* d_ws, size_t ws_size,
                              hipStream_t stream) {
  const float* u = (const float*)d_in[0];      // (4, 8192, 1024) f32
  const float* W = (const float*)d_in[3];      // (3072, 1024) f32
  float* out = (float*)d_out;                  // (4, 8192, 1024) f32

  char* ws = (char*)d_ws;
  float* G   = (float*)ws;                               // 4*1024*1024 f32 = 16 MB
  float* kvb = (float*)(ws + (size_t)16 * 1024 * 1024);  // 4*16*64*64  f32 =  1 MB
  float* M   = (float*)(ws + (size_t)17 * 1024 * 1024);  // 4*1024*1024 f32 = 16 MB

  gram_kernel<<<dim3(256),  dim3(256), 0, stream>>>(u, G);
  kv_kernel  <<<dim3(64),   dim3(256), 0, stream>>>(W, G, kvb);
  mcat_kernel<<<dim3(512),  dim3(256), 0, stream>>>(W, kvb, M);
  y_kernel   <<<dim3(2048), dim3(256), 0, stream>>>(u, M, out);
}